// polyHype_53145925320941
// MI455X (gfx1250) — compile-verified
//
#include <hip/hip_runtime.h>
#include <hip/hip_bf16.h>
#include <math.h>

// ---------------------------------------------------------------------------
// Fused PathCon-style hypergraph aggregation + classifier for MI455X (gfx1250)
//
//  - per-(row,slot) type histogram in LDS (lane-exclusive bins, no atomics)
//  - node embedding mean in LDS
//  - three small GEMMs done with V_WMMA_F32_16X16X4_F32 (exact f32 math):
//      tmp    = agg_pre  @ type_features        (16x32 @ 32x32)
//      scores = (tmp+self) @ W1 + b1            (16x32 @ 32x32)
//      nfeat  = node_mean @ W2 + b2             (16x64 @ 64x32)
// ---------------------------------------------------------------------------

typedef __attribute__((ext_vector_type(2))) float v2f;
typedef __attribute__((ext_vector_type(8))) float v8f;

#define WAVES 2      // waves per block; each wave owns a 16-row tile
#define TDIM 32      // n_types == output dim
#define EDIM 64      // nodeEmb width
#define HSZ  4       // hedge slots per row
#define NSS  64      // neighbor samples per slot

// A-fragment for V_WMMA_F32_16X16X4_F32: 16x4 f32.
// lane L: M = L%16 ; VGPR0 holds K = k0 + (L<16?0:2), VGPR1 holds K+1.
__device__ __forceinline__ v2f load_a_frag(const float* A, int lda, int k0, int lane) {
    const int m  = lane & 15;
    const int kh = (lane >> 4) << 1;          // 0 for lanes 0-15, 2 for 16-31
    v2f a;
    a.x = A[m * lda + k0 + kh];
    a.y = A[m * lda + k0 + kh + 1];
    return a;
}

// B-fragment: 4x16 f32, row-striped across lanes (N = lane%16), K split like A.
__device__ __forceinline__ v2f load_b_frag(const float* Bm, int ldb, int k0, int n0, int lane) {
    const int n  = lane & 15;
    const int kh = (lane >> 4) << 1;
    v2f b;
    b.x = Bm[(k0 + kh)     * ldb + n0 + n];
    b.y = Bm[(k0 + kh + 1) * ldb + n0 + n];
    return b;
}

__global__ __launch_bounds__(WAVES * 32) void polyhype_fused_kernel(
    const int*   __restrict__ neighbors,       // [B, HSZ]
    const int*   __restrict__ train_hedges,    // [B]
    const int*   __restrict__ labels,          // [B]
    const int*   __restrict__ neighborhedges,  // [N_NODES, NSS]
    const int*   __restrict__ hedgetypes,      // [N_HEDGES]
    const int*   __restrict__ nodeEmb,         // [N_NODES, EDIM]
    const float* __restrict__ TF,              // [TDIM, TDIM]
    const float* __restrict__ W1,              // [TDIM, TDIM]
    const float* __restrict__ b1,              // [TDIM]
    const float* __restrict__ W2,              // [EDIM, TDIM]
    const float* __restrict__ b2,              // [TDIM]
    float*       __restrict__ out_scores,      // [B, TDIM]
    float*       __restrict__ out_norm,        // [B, TDIM]
    float*       __restrict__ out_emb)         // [B, 2*TDIM]
{
    __shared__ int   s_cnt[WAVES][64][TDIM];     // [wave][row*4+slot][type]
    __shared__ int   s_den[WAVES][64];
    __shared__ float s_nmean[WAVES][16][EDIM];
    __shared__ float s_agg[WAVES][16][TDIM];
    __shared__ float s_tmp[WAVES][16][TDIM];

    const int lane    = threadIdx.x & 31;
    const int wave    = threadIdx.x >> 5;
    const int rowBase = (blockIdx.x * WAVES + wave) * 16;

    // ---- zero histogram bins -------------------------------------------------
    {
        int* cw = &s_cnt[wave][0][0];
        #pragma unroll
        for (int i = 0; i < 64; ++i) cw[i * TDIM + lane] = 0;
        s_den[wave][lane]      = 0;
        s_den[wave][lane + 32] = 0;
    }
    __syncthreads();

    // ---- masked per-type histogram: each lane owns 2 (row,slot) pairs --------
    #pragma unroll
    for (int pp = 0; pp < 2; ++pp) {
        const int p    = lane + pp * 32;           // pair id 0..63
        const int r    = p >> 2;
        const int h    = p & 3;
        const int brow = rowBase + r;
        const int node = neighbors[brow * HSZ + h];
        const int th   = train_hedges[brow];
        const int4* rowp = (const int4*)(neighborhedges + (size_t)node * NSS);
        int* cp = &s_cnt[wave][p][0];
        int den = 0;
        #pragma unroll 4
        for (int j = 0; j < NSS / 4; ++j) {        // global_load_b128 per step
            const int4 e = rowp[j];
            if (e.x != th) { cp[hedgetypes[e.x]] += 1; ++den; }
            if (e.y != th) { cp[hedgetypes[e.y]] += 1; ++den; }
            if (e.z != th) { cp[hedgetypes[e.z]] += 1; ++den; }
            if (e.w != th) { cp[hedgetypes[e.w]] += 1; ++den; }
        }
        s_den[wave][p] = den;
    }

    // ---- node embedding mean over the 4 hedge slots --------------------------
    for (int r = 0; r < 16; ++r) {
        const int brow = rowBase + r;
        const int n0 = neighbors[brow * HSZ + 0];
        const int n1 = neighbors[brow * HSZ + 1];
        const int n2 = neighbors[brow * HSZ + 2];
        const int n3 = neighbors[brow * HSZ + 3];
        #pragma unroll
        for (int q = 0; q < 2; ++q) {
            const int e = lane + q * 32;
            const int s = nodeEmb[(size_t)n0 * EDIM + e] + nodeEmb[(size_t)n1 * EDIM + e]
                        + nodeEmb[(size_t)n2 * EDIM + e] + nodeEmb[(size_t)n3 * EDIM + e];
            s_nmean[wave][r][e] = (float)s * 0.25f;
        }
    }
    __syncthreads();

    // ---- per-type masked-mean weights: agg_pre = mean_h(count/max(den,1)) ----
    for (int r = 0; r < 16; ++r) {
        float s = 0.0f;
        #pragma unroll
        for (int h = 0; h < HSZ; ++h) {
            const float d = fmaxf((float)s_den[wave][r * HSZ + h], 1.0f);
            s += (float)s_cnt[wave][r * HSZ + h][lane] / d;
        }
        s_agg[wave][r][lane] = 0.25f * s;          // lane == type index
    }
    __syncthreads();

    // ---- GEMM1 (WMMA): tmp = agg_pre @ TF ------------------------------------
    #pragma unroll
    for (int nt = 0; nt < 2; ++nt) {
        v8f c = {};
        #pragma unroll
        for (int k0 = 0; k0 < TDIM; k0 += 4) {
            v2f a   = load_a_frag(&s_agg[wave][0][0], TDIM, k0, lane);
            v2f bfr = load_b_frag(TF, TDIM, k0, nt * 16, lane);
            c = __builtin_amdgcn_wmma_f32_16x16x4_f32(false, a, false, bfr,
                                                      (short)0, c, false, false);
        }
        const int n  = (lane & 15) + nt * 16;
        const int mb = (lane >> 4) << 3;           // 0 or 8
        #pragma unroll
        for (int r2 = 0; r2 < 8; ++r2) s_tmp[wave][mb + r2][n] = c[r2];
    }
    __syncthreads();

    // ---- add self one-hot vector: tmp[r] += TF[labels[r]] --------------------
    for (int r = 0; r < 16; ++r) {
        const int lab = labels[rowBase + r];
        s_tmp[wave][r][lane] += TF[lab * TDIM + lane];
    }
    __syncthreads();

    // ---- GEMM2 (WMMA): scores = tmp @ W1 + b1 ; GEMM3: nfeat = nmean @ W2 + b2
    #pragma unroll
    for (int nt = 0; nt < 2; ++nt) {
        const int n  = (lane & 15) + nt * 16;
        const int mb = (lane >> 4) << 3;

        v8f c;
        const float bias1 = b1[n];
        #pragma unroll
        for (int r2 = 0; r2 < 8; ++r2) c[r2] = bias1;
        #pragma unroll
        for (int k0 = 0; k0 < TDIM; k0 += 4) {
            v2f a   = load_a_frag(&s_tmp[wave][0][0], TDIM, k0, lane);
            v2f bfr = load_b_frag(W1, TDIM, k0, nt * 16, lane);
            c = __builtin_amdgcn_wmma_f32_16x16x4_f32(false, a, false, bfr,
                                                      (short)0, c, false, false);
        }
        #pragma unroll
        for (int r2 = 0; r2 < 8; ++r2) {
            const int row = rowBase + mb + r2;
            const float v = c[r2];
            out_scores[row * TDIM + n]    = v;
            out_norm[row * TDIM + n]      = 1.0f / (1.0f + expf(-v));
            out_emb[row * 2 * TDIM + n]   = v;
        }

        v8f c2;
        const float bias2 = b2[n];
        #pragma unroll
        for (int r2 = 0; r2 < 8; ++r2) c2[r2] = bias2;
        #pragma unroll
        for (int k0 = 0; k0 < EDIM; k0 += 4) {
            v2f a   = load_a_frag(&s_nmean[wave][0][0], EDIM, k0, lane);
            v2f bfr = load_b_frag(W2, TDIM, k0, nt * 16, lane);
            c2 = __builtin_amdgcn_wmma_f32_16x16x4_f32(false, a, false, bfr,
                                                       (short)0, c2, false, false);
        }
        #pragma unroll
        for (int r2 = 0; r2 < 8; ++r2) {
            const int row = rowBase + mb + r2;
            out_emb[row * 2 * TDIM + TDIM + n] = c2[r2];
        }
    }
}

extern "C" void kernel_launch(void* const* d_in, const int* in_sizes, int n_in,
                              void* d_out, int out_size, void* d_ws, size_t ws_size,
                              hipStream_t stream) {
    const int*   neighbors      = (const int*)  d_in[0];
    const int*   train_hedges   = (const int*)  d_in[1];
    const int*   labels         = (const int*)  d_in[2];
    const int*   neighborhedges = (const int*)  d_in[3];
    const int*   hedgetypes     = (const int*)  d_in[4];
    const int*   nodeEmb        = (const int*)  d_in[5];
    const float* TF             = (const float*)d_in[6];
    const float* W1             = (const float*)d_in[7];
    const float* b1             = (const float*)d_in[8];
    const float* W2             = (const float*)d_in[9];
    const float* b2             = (const float*)d_in[10];

    const int B = in_sizes[1];                  // train_hedges length == batch
    float* out        = (float*)d_out;
    float* out_scores = out;
    float* out_norm   = out + (size_t)B * TDIM;
    float* out_emb    = out + (size_t)B * 2 * TDIM;

    const int rowsPerBlock = WAVES * 16;
    dim3 grid((B + rowsPerBlock - 1) / rowsPerBlock);
    dim3 block(WAVES * 32);
    polyhype_fused_kernel<<<grid, block, 0, stream>>>(
        neighbors, train_hedges, labels, neighborhedges, hedgetypes, nodeEmb,
        TF, W1, b1, W2, b2, out_scores, out_norm, out_emb);
}